// DeformAttn_14937896256238
// MI455X (gfx1250) — compile-verified
//
#include <hip/hip_runtime.h>
#include <hip/hip_bf16.h>

// Deformable attention forward for MI455X (gfx1250), wave32 + WMMA f32 16x16x4.
// Pipeline: 3 input GEMMs -> build padded point vectors -> softmax(P=4) ->
// WMMA-based fused distance+argmin KNN -> gather/weighted-sum -> output GEMM.
// Register-blocked GEMM (32x64 per wave); KNN handles 64 queries/wave with a
// software-pipelined (double-buffered) candidate stream.

typedef __attribute__((ext_vector_type(2))) float v2f;
typedef __attribute__((ext_vector_type(8))) float v8f;

#define BB   2
#define LQ   1024
#define LIN  4096
#define CC   256
#define MH   8
#define NP   4
#define DH   32
#define QT   4

// ---------------------------------------------------------------------------
// Register-blocked f32 WMMA GEMM: out[M,N] = X[M,K] @ W[N,K]^T + bias[N]
// One wave computes a (16*MT) x (16*NT) output block.
// grid = (M/(16*MT), N/(16*NT)).
// ---------------------------------------------------------------------------
template <int MT, int NT>
__global__ void __launch_bounds__(32)
wmma_gemm_f32(const float* __restrict__ X, const float* __restrict__ W,
              const float* __restrict__ bias, float* __restrict__ out,
              int Ncols, int K) {
    const int lane = threadIdx.x;
    const int l15  = lane & 15;
    const int kh   = lane >> 4;           // 0: k,k+1   1: k+2,k+3
    const int row0 = blockIdx.x * (16 * MT);
    const int col0 = blockIdx.y * (16 * NT);

    const float* __restrict__ xr[MT];
    const float* __restrict__ wr[NT];
#pragma unroll
    for (int mt = 0; mt < MT; ++mt)
        xr[mt] = X + (size_t)(row0 + mt * 16 + l15) * K + 2 * kh;
#pragma unroll
    for (int nt = 0; nt < NT; ++nt)
        wr[nt] = W + (size_t)(col0 + nt * 16 + l15) * K + 2 * kh;

    v8f acc[MT][NT];
#pragma unroll
    for (int mt = 0; mt < MT; ++mt)
#pragma unroll
        for (int nt = 0; nt < NT; ++nt) {
            v8f z = {};
            acc[mt][nt] = z;
        }

#pragma unroll 2
    for (int k = 0; k < K; k += 4) {
        v2f a[MT], b[NT];
#pragma unroll
        for (int mt = 0; mt < MT; ++mt) a[mt] = *(const v2f*)(xr[mt] + k);
#pragma unroll
        for (int nt = 0; nt < NT; ++nt) b[nt] = *(const v2f*)(wr[nt] + k);
#pragma unroll
        for (int mt = 0; mt < MT; ++mt)
#pragma unroll
            for (int nt = 0; nt < NT; ++nt)
                acc[mt][nt] = __builtin_amdgcn_wmma_f32_16x16x4_f32(
                    /*neg_a=*/false, a[mt], /*neg_b=*/false, b[nt],
                    /*c_mod=*/(short)0, acc[mt][nt],
                    /*reuse_a=*/false, /*reuse_b=*/false);
    }

#pragma unroll
    for (int nt = 0; nt < NT; ++nt) {
        const int   col  = col0 + nt * 16;
        const float bcol = bias[col + l15];
#pragma unroll
        for (int mt = 0; mt < MT; ++mt) {
            const int row = row0 + mt * 16;
#pragma unroll
            for (int r = 0; r < 8; ++r) {
                // C/D layout: VGPR r -> M = r + 8*kh, N = lane&15
                out[(size_t)(row + r + 8 * kh) * Ncols + col + l15] =
                    acc[mt][nt][r] + bcol;
            }
        }
    }
}

// ---------------------------------------------------------------------------
// Build padded 4-vectors for the rank-4 distance trick.
//   S4[n] = [sx, sy, sz, 1]           (sampling loc = query_point + offset)
//   V4[l] = [-2vx, -2vy, -2vz, |v|^2] (candidate value point)
// ---------------------------------------------------------------------------
__global__ void build_pts(const float* __restrict__ off,
                          const float* __restrict__ qpts,
                          const float* __restrict__ ipts,
                          float* __restrict__ S4, float* __restrict__ V4) {
    const int i = blockIdx.x * blockDim.x + threadIdx.x;
    if (i < BB * LQ * MH * NP) {
        const int b  = i >> 15;            // / (LQ*MH*NP)
        const int q  = (i >> 5) & (LQ - 1);
        const int mp = i & 31;
        const float* o = off  + ((size_t)(b * LQ + q)) * (MH * NP * 3) + mp * 3;
        const float* p = qpts + ((size_t)(b * LQ + q)) * 3;
        float4 s;
        s.x = p[0] + o[0];
        s.y = p[1] + o[1];
        s.z = p[2] + o[2];
        s.w = 1.0f;
        *(float4*)(S4 + (size_t)i * 4) = s;
    }
    if (i < BB * LIN) {
        const float* v = ipts + (size_t)i * 3;
        const float x = v[0], y = v[1], z = v[2];
        float4 t;
        t.x = -2.0f * x;
        t.y = -2.0f * y;
        t.z = -2.0f * z;
        t.w = x * x + y * y + z * z;
        *(float4*)(V4 + (size_t)i * 4) = t;
    }
}

// ---------------------------------------------------------------------------
// Softmax over P=4 per (b, q, m) row, in place.
// ---------------------------------------------------------------------------
__global__ void softmax_p4(float* __restrict__ a, int rows) {
    const int r = blockIdx.x * blockDim.x + threadIdx.x;
    if (r >= rows) return;
    float4 v = *(float4*)(a + (size_t)r * 4);
    const float mx = fmaxf(fmaxf(v.x, v.y), fmaxf(v.z, v.w));
    const float e0 = __expf(v.x - mx), e1 = __expf(v.y - mx);
    const float e2 = __expf(v.z - mx), e3 = __expf(v.w - mx);
    const float inv = 1.0f / (e0 + e1 + e2 + e3);
    v.x = e0 * inv; v.y = e1 * inv; v.z = e2 * inv; v.w = e3 * inv;
    *(float4*)(a + (size_t)r * 4) = v;
}

// ---------------------------------------------------------------------------
// Fused distance + argmin KNN via WMMA f32 16x16x4.
// One wave handles QT*16 = 64 sampling points; each candidate A-fragment is
// reused across QT WMMAs. D[m][n] = |v_m|^2 - 2 * s_n . v_m
// (A = candidates along M, B = queries along N). Candidate stream is
// double-buffered so the next load overlaps the argmin VALU chain.
// ---------------------------------------------------------------------------
__device__ __forceinline__ void knn_step(const v2f a, const v2f* bq, int cib,
                                         float* best, int* bidx) {
#pragma unroll
    for (int t = 0; t < QT; ++t) {
        v8f c = {};
        c = __builtin_amdgcn_wmma_f32_16x16x4_f32(
            false, a, false, bq[t], (short)0, c, false, false);
#pragma unroll
        for (int r = 0; r < 8; ++r) {
            const float dv = c[r];
            const int   ci = cib + r;
            if (dv < best[t]) { best[t] = dv; bidx[t] = ci; }
        }
    }
}

__global__ void __launch_bounds__(32)
knn_wmma(const float* __restrict__ S4, const float* __restrict__ V4,
         int* __restrict__ idxOut) {
    const int lane = threadIdx.x;
    const int l15  = lane & 15;
    const int kh   = lane >> 4;
    const int n0   = blockIdx.x * (16 * QT);       // sampling-point base
    const int b    = n0 >> 15;                     // / (LQ*MH*NP)

    // Query fragments (B matrix, 4x16 each), loop-invariant.
    v2f bq[QT];
#pragma unroll
    for (int t = 0; t < QT; ++t)
        bq[t] = *(const v2f*)(S4 + (size_t)(n0 + t * 16 + l15) * 4 + 2 * kh);

    const float* __restrict__ vptr = V4 + (size_t)b * LIN * 4
                                        + (size_t)l15 * 4 + 2 * kh;

    float best[QT];
    int   bidx[QT];
#pragma unroll
    for (int t = 0; t < QT; ++t) { best[t] = 3.4e38f; bidx[t] = 0; }

    v2f a_cur = *(const v2f*)(vptr);
    for (int cb = 0; cb < LIN - 16; cb += 16) {
        const v2f a_next = *(const v2f*)(vptr + (size_t)(cb + 16) * 4);
        knn_step(a_cur, bq, cb + (kh << 3), best, bidx);
        a_cur = a_next;
    }
    knn_step(a_cur, bq, (LIN - 16) + (kh << 3), best, bidx);

    // Merge the two lane-halves (same query, disjoint candidate rows).
#pragma unroll
    for (int t = 0; t < QT; ++t) {
        const float obest = __shfl_xor(best[t], 16, 32);
        const int   oidx  = __shfl_xor(bidx[t], 16, 32);
        if (obest < best[t] || (obest == best[t] && oidx < bidx[t])) {
            best[t] = obest; bidx[t] = oidx;
        }
        if (lane < 16) idxOut[n0 + t * 16 + l15] = bidx[t];
    }
}

// ---------------------------------------------------------------------------
// Gather nearest value rows and reduce over P with attention weights.
// out_tmp[b,q,c] = sum_p attw[b,q,m,p] * value[b, idx[b,q,m,p], c],  m = c/32
// ---------------------------------------------------------------------------
__global__ void __launch_bounds__(CC)
gather_weighted(const float* __restrict__ value, const int* __restrict__ idx,
                const float* __restrict__ attw, float* __restrict__ out_tmp) {
    const int bq = blockIdx.x;        // 0 .. B*LQ-1
    const int c  = threadIdx.x;       // 0 .. 255
    const int b  = bq >> 10;
    const int m  = c >> 5;
    const int*   id = idx  + ((size_t)bq * MH + m) * NP;
    const float* aw = attw + ((size_t)bq * MH + m) * NP;
    const float* vb = value + (size_t)b * LIN * CC;
    float acc = 0.0f;
#pragma unroll
    for (int p = 0; p < NP; ++p)
        acc += aw[p] * vb[(size_t)id[p] * CC + c];
    out_tmp[(size_t)bq * CC + c] = acc;
}

// ---------------------------------------------------------------------------
extern "C" void kernel_launch(void* const* d_in, const int* in_sizes, int n_in,
                              void* d_out, int out_size, void* d_ws, size_t ws_size,
                              hipStream_t stream) {
    const float* query = (const float*)d_in[0];
    const float* qpts  = (const float*)d_in[1];
    const float* inp   = (const float*)d_in[2];
    const float* ipts  = (const float*)d_in[3];
    const float* Wv    = (const float*)d_in[4];
    const float* bv    = (const float*)d_in[5];
    const float* Ws    = (const float*)d_in[6];
    const float* bs    = (const float*)d_in[7];
    const float* Wa    = (const float*)d_in[8];
    const float* ba    = (const float*)d_in[9];
    const float* Wo    = (const float*)d_in[10];
    const float* bo    = (const float*)d_in[11];
    float* out = (float*)d_out;

    // Workspace layout (all 256B-aligned by construction).
    char* w = (char*)d_ws;
    float* value = (float*)w;  w += (size_t)BB * LIN * CC * 4;            // 8 MB
    float* off   = (float*)w;  w += (size_t)BB * LQ * MH * NP * 3 * 4;    // 768 KB
    float* attw  = (float*)w;  w += (size_t)BB * LQ * MH * NP * 4;        // 256 KB
    float* S4    = (float*)w;  w += (size_t)BB * LQ * MH * NP * 4 * 4;    // 1 MB
    float* V4    = (float*)w;  w += (size_t)BB * LIN * 4 * 4;             // 128 KB
    int*   idx   = (int*)w;    w += (size_t)BB * LQ * MH * NP * 4;        // 256 KB
    float* otmp  = (float*)w;  w += (size_t)BB * LQ * CC * 4;             // 2 MB

    const dim3 wave(32);

    // 1) value = inp @ Wv^T + bv      (8192 x 256) x (256 x 256), 32x64/wave
    wmma_gemm_f32<2, 4><<<dim3((BB * LIN) / 32, CC / 64), wave, 0, stream>>>(
        inp, Wv, bv, value, CC, CC);

    // 2) off = query @ Ws^T + bs      (2048 x 96), 32x32/wave
    wmma_gemm_f32<2, 2><<<dim3((BB * LQ) / 32, (MH * NP * 3) / 32), wave, 0, stream>>>(
        query, Ws, bs, off, MH * NP * 3, CC);

    // 3) attn logits = query @ Wa^T + ba   (2048 x 32) -> attw buffer
    wmma_gemm_f32<2, 2><<<dim3((BB * LQ) / 32, (MH * NP) / 32), wave, 0, stream>>>(
        query, Wa, ba, attw, MH * NP, CC);

    // 4) padded point vectors for the rank-4 distance trick
    {
        const int n = BB * LQ * MH * NP;   // 65536 (covers BB*LIN too)
        build_pts<<<(n + 255) / 256, 256, 0, stream>>>(off, qpts, ipts, S4, V4);
    }

    // 5) softmax over P=4
    {
        const int rows = BB * LQ * MH;     // 16384
        softmax_p4<<<(rows + 255) / 256, 256, 0, stream>>>(attw, rows);
    }

    // 6) fused distance + argmin KNN (WMMA), 64 sampling points per wave
    knn_wmma<<<(BB * LQ * MH * NP) / (16 * QT), wave, 0, stream>>>(S4, V4, idx);

    // 7) gather + weighted sum over P -> (B, Lq, C)
    gather_weighted<<<BB * LQ, CC, 0, stream>>>(value, idx, attw, otmp);

    // 8) out = otmp @ Wo^T + bo
    wmma_gemm_f32<2, 4><<<dim3((BB * LQ) / 32, CC / 64), wave, 0, stream>>>(
        otmp, Wo, bo, out, CC, CC);
}